// SubgraphPredictor_73684458930835
// MI455X (gfx1250) — compile-verified
//
#include <hip/hip_runtime.h>
#include <math.h>

// MeshGraphNets-style forward for MI455X (gfx1250, wave32).
// All 128x128-K GEMMs run on matrix units via V_WMMA_F32_16X16X4_F32 (fp32:
// workload is HBM/L2-bound, so fp32 matrix math is free accuracy).
// Each workgroup now processes 4 row-tiles (64 rows): weight (B) fragments are
// reused 4x (L2 traffic /4) and the 4 independent accumulator chains keep the
// XDL pipe busy (no serial RAW on a single accumulator).

typedef __attribute__((ext_vector_type(2))) float v2f;
typedef __attribute__((ext_vector_type(8))) float v8f;

#define AS_LD 132   // padded LDS row stride (floats): bank-conflict-free frags
#define TILES 4     // 16-row tiles per workgroup (64 rows)

// ---------------------------------------------------------------------------
// Generic fused GEMM:  Y[M,128] = act( sum_i gather(X_i)[M,128] @ W_i + bias )
//                      (+ residual R).  W stacked row-major: W[i*16384+k*128+n]
// Workgroup = 64 rows (4 tiles); wave w computes columns [16w,16w+16), K=128.
// ---------------------------------------------------------------------------
template <int NIN, bool RELU, bool RESID>
__global__ __launch_bounds__(256) void gemm128_wmma(
    const float* __restrict__ X0, const int* __restrict__ g0,
    const float* __restrict__ X1, const int* __restrict__ g1,
    const float* __restrict__ X2, const int* __restrict__ g2,
    const float* __restrict__ W, const float* __restrict__ bias,
    const float* __restrict__ Rm, float* __restrict__ Y, int M) {
  __shared__ float As[NIN * TILES * 16 * AS_LD];

  const int row0 = blockIdx.x * (16 * TILES);
  const int tid = threadIdx.x;

  // Stage A tiles (TILES x 16 rows x 128 K) into LDS.
  {
    const int lr = tid >> 4;          // local row 0..15
    const int c0 = (tid & 15) * 8;    // col base
    const float* Xs[3] = {X0, X1, X2};
    const int* gs[3] = {g0, g1, g2};
#pragma unroll
    for (int t = 0; t < TILES; ++t) {
      int gr = row0 + t * 16 + lr;
      if (gr >= M) gr = M - 1;        // clamp: keeps addresses valid
#pragma unroll
      for (int i = 0; i < NIN; ++i) {
        long srow = gs[i] ? (long)gs[i][gr] : (long)gr;
        const float4* src = (const float4*)(Xs[i] + srow * 128 + c0);
        float4 a0 = src[0];
        float4 a1 = src[1];
        float4* dst =
            (float4*)&As[((i * TILES + t) * 16 + lr) * AS_LD + c0];
        dst[0] = a0;
        dst[1] = a1;
      }
    }
  }
  __syncthreads();

  const int lane = tid & 31;
  const int wave = tid >> 5;
  const int lh = lane & 15;
  const int hi = lane >> 4;          // which half of the 4-wide K chunk
  const int col = wave * 16 + lh;    // output column for this lane

  const float bv = bias[col];
  v8f acc[TILES];
#pragma unroll
  for (int t = 0; t < TILES; ++t)
    acc[t] = (v8f){bv, bv, bv, bv, bv, bv, bv, bv};

#pragma unroll
  for (int i = 0; i < NIN; ++i) {
    const float* Wi = W + (long)i * 128 * 128 + col;
#pragma unroll
    for (int kk = 0; kk < 32; ++kk) {
      const int k0 = kk * 4 + hi * 2;
      // B frag (4x16 f32): element (k, n=col); shared by all 4 row-tiles.
      v2f b;
      b.x = Wi[(long)k0 * 128];
      b.y = Wi[(long)(k0 + 1) * 128];
#pragma unroll
      for (int t = 0; t < TILES; ++t) {
        // A frag (16x4 f32): lanes 0-15 hold K=0,1; lanes 16-31 hold K=2,3.
        v2f a = *(const v2f*)(&As[((i * TILES + t) * 16 + lh) * AS_LD + k0]);
        acc[t] = __builtin_amdgcn_wmma_f32_16x16x4_f32(
            false, a, false, b, (short)0, acc[t], false, false);
      }
    }
  }

  // C/D layout: VGPR v, lanes 0-15 -> M=v; lanes 16-31 -> M=v+8.
#pragma unroll
  for (int t = 0; t < TILES; ++t) {
#pragma unroll
    for (int v = 0; v < 8; ++v) {
      const int grow = row0 + t * 16 + v + hi * 8;
      if (grow < M) {
        float val = acc[t][v];
        if (RELU) val = fmaxf(val, 0.f);
        const long idx = (long)grow * 128 + col;
        if (RESID) val += Rm[idx];
        Y[idx] = val;
      }
    }
  }
}

// ---------------------------------------------------------------------------
// Elementwise / reduction kernels
// ---------------------------------------------------------------------------
__global__ void fzero_kernel(float* __restrict__ p, long n) {
  long i = (long)blockIdx.x * blockDim.x + threadIdx.x;
  const long stride = (long)gridDim.x * blockDim.x;
  for (; i < n; i += stride) p[i] = 0.f;
}

// edge raw features (rel, |rel|) + partial sums for edge stats -> stats[0..5]
__global__ __launch_bounds__(256) void edge_raw_stats_kernel(
    const float* __restrict__ pos, const int* __restrict__ snd,
    const int* __restrict__ rcv, float* __restrict__ ef_raw,
    float* __restrict__ stats, int E) {
  __shared__ float red[256];
  float s0 = 0.f, s1 = 0.f, s2 = 0.f, q0 = 0.f, q1 = 0.f, q2 = 0.f;
  int i = blockIdx.x * blockDim.x + threadIdx.x;
  const int stride = gridDim.x * blockDim.x;
  for (int e = i; e < E; e += stride) {
    const int a = snd[e], b = rcv[e];
    const float rx = pos[2 * a] - pos[2 * b];
    const float ry = pos[2 * a + 1] - pos[2 * b + 1];
    const float nr = sqrtf(rx * rx + ry * ry);
    ef_raw[3 * e] = rx;
    ef_raw[3 * e + 1] = ry;
    ef_raw[3 * e + 2] = nr;
    s0 += rx; s1 += ry; s2 += nr;
    q0 += rx * rx; q1 += ry * ry; q2 += nr * nr;
  }
  float vals[6] = {s0, s1, s2, q0, q1, q2};
  for (int t = 0; t < 6; ++t) {
    red[threadIdx.x] = vals[t];
    __syncthreads();
    for (int o = 128; o > 0; o >>= 1) {
      if (threadIdx.x < (unsigned)o) red[threadIdx.x] += red[threadIdx.x + o];
      __syncthreads();
    }
    if (threadIdx.x == 0) atomicAdd(&stats[t], red[0]);
    __syncthreads();
  }
}

// node stats over zl -> stats[6..11]; target=(zt-zl) stats -> stats[12..17]
__global__ __launch_bounds__(256) void node_stats_kernel(
    const float* __restrict__ zl, const float* __restrict__ zt,
    float* __restrict__ stats, int N) {
  __shared__ float red[256];
  float vals[12];
  for (int t = 0; t < 12; ++t) vals[t] = 0.f;
  int i = blockIdx.x * blockDim.x + threadIdx.x;
  const int stride = gridDim.x * blockDim.x;
  for (int n = i; n < N; n += stride) {
    for (int c = 0; c < 3; ++c) {
      const float zv = zl[3 * n + c];
      const float tv = zt[3 * n + c] - zv;
      vals[c] += zv;
      vals[3 + c] += zv * zv;
      vals[6 + c] += tv;
      vals[9 + c] += tv * tv;
    }
  }
  for (int t = 0; t < 12; ++t) {
    red[threadIdx.x] = vals[t];
    __syncthreads();
    for (int o = 128; o > 0; o >>= 1) {
      if (threadIdx.x < (unsigned)o) red[threadIdx.x] += red[threadIdx.x + o];
      __syncthreads();
    }
    if (threadIdx.x == 0) atomicAdd(&stats[6 + t], red[0]);
    __syncthreads();
  }
}

// sums -> mean/std:  edge(m:18 s:21)  node(m:24 s:27)  target(m:30 s:33)
__global__ void finalize_stats_kernel(float* __restrict__ stats, int E, int N) {
  const int i = threadIdx.x;
  if (i < 3) {
    const float m = stats[i] / (float)E;
    const float v = fmaxf(stats[3 + i] / (float)E - m * m, 0.f);
    stats[18 + i] = m;
    stats[21 + i] = fmaxf(sqrtf(v), 1e-8f);
  } else if (i < 6) {
    const int c = i - 3;
    const float m = stats[6 + c] / (float)N;
    const float v = fmaxf(stats[9 + c] / (float)N - m * m, 0.f);
    stats[24 + c] = m;
    stats[27 + c] = fmaxf(sqrtf(v), 1e-8f);
  } else if (i < 9) {
    const int c = i - 6;
    const float m = stats[12 + c] / (float)N;
    const float v = fmaxf(stats[15 + c] / (float)N - m * m, 0.f);
    stats[30 + c] = m;
    stats[33 + c] = fmaxf(sqrtf(v), 1e-8f);
  }
}

// encoder layer 1 (K=3, done on VALU): h = relu(((x-m)/s) @ W1[3,128] + b1)
__global__ void enc_l1_kernel(const float* __restrict__ x3,
                              const float* __restrict__ m,
                              const float* __restrict__ s,
                              const float* __restrict__ W1,
                              const float* __restrict__ b1,
                              float* __restrict__ h, int M) {
  long i = (long)blockIdx.x * blockDim.x + threadIdx.x;
  const long stride = (long)gridDim.x * blockDim.x;
  const long total = (long)M * 128;
  for (; i < total; i += stride) {
    const long r = i >> 7;
    const int j = (int)(i & 127);
    const float x0 = (x3[3 * r] - m[0]) / s[0];
    const float x1 = (x3[3 * r + 1] - m[1]) / s[1];
    const float x2 = (x3[3 * r + 2] - m[2]) / s[2];
    const float v = b1[j] + x0 * W1[j] + x1 * W1[128 + j] + x2 * W1[256 + j];
    h[i] = fmaxf(v, 0.f);
  }
}

// segment_sum scatter:  agg[rcv[e], j..j+3] += ef[e, j..j+3]   (4-wide)
__global__ void scatter_add_kernel(const float* __restrict__ ef,
                                   const int* __restrict__ rcv,
                                   float* __restrict__ agg, int E) {
  long i = (long)blockIdx.x * blockDim.x + threadIdx.x;
  const long stride = (long)gridDim.x * blockDim.x;
  const long total = (long)E * 32;  // 32 quads per edge row
  for (; i < total; i += stride) {
    const long e = i >> 5;
    const int j4 = (int)(i & 31) * 4;
    const long src = e * 128 + j4;
    float* dst = &agg[(long)rcv[e] * 128 + j4];
    const float4 v = *(const float4*)(ef + src);
    atomicAdd(dst + 0, v.x);
    atomicAdd(dst + 1, v.y);
    atomicAdd(dst + 2, v.z);
    atomicAdd(dst + 3, v.w);
  }
}

// decoder layer 2 (Ncols=3, VALU): out[n,c] = h[n,:] @ W2[:,c] + b2[c]
__global__ void dec_l2_kernel(const float* __restrict__ h,
                              const float* __restrict__ W2,
                              const float* __restrict__ b2,
                              float* __restrict__ out, int M) {
  long i = (long)blockIdx.x * blockDim.x + threadIdx.x;
  const long stride = (long)gridDim.x * blockDim.x;
  const long total = (long)M * 3;
  for (; i < total; i += stride) {
    const long r = i / 3;
    const int c = (int)(i % 3);
    const float* hr = h + r * 128;
    float acc = b2[c];
    for (int k = 0; k < 128; ++k) acc += hr[k] * W2[k * 3 + c];
    out[i] = acc;
  }
}

// loss partials + z_pred + per-column squared error -> stats[36..39]
__global__ __launch_bounds__(256) void loss_pred_kernel(
    const float* __restrict__ zl, const float* __restrict__ zt,
    const float* __restrict__ out, const float* __restrict__ stats,
    float* __restrict__ zpred, float* __restrict__ acc_out, int N) {
  __shared__ float red[256];
  const float* om = stats + 30;
  const float* os = stats + 33;
  float lsum = 0.f, c0 = 0.f, c1 = 0.f, c2 = 0.f;
  long i = (long)blockIdx.x * blockDim.x + threadIdx.x;
  const long stride = (long)gridDim.x * blockDim.x;
  const long total = (long)N * 3;
  for (; i < total; i += stride) {
    const int c = (int)(i % 3);
    const float t = zt[i] - zl[i];
    const float tn = (t - om[c]) / os[c];
    const float o = out[i];
    const float d = tn - o;
    lsum += d * d;
    const float zp = zl[i] + o * os[c] + om[c];
    zpred[i] = zp;
    const float e = zp - zt[i];
    const float e2 = e * e;
    if (c == 0) c0 += e2; else if (c == 1) c1 += e2; else c2 += e2;
  }
  float vals[4] = {lsum, c0, c1, c2};
  for (int t = 0; t < 4; ++t) {
    red[threadIdx.x] = vals[t];
    __syncthreads();
    for (int o = 128; o > 0; o >>= 1) {
      if (threadIdx.x < (unsigned)o) red[threadIdx.x] += red[threadIdx.x + o];
      __syncthreads();
    }
    if (threadIdx.x == 0) atomicAdd(&acc_out[36 + t], red[0]);
    __syncthreads();
  }
}

__global__ void final_scalars_kernel(const float* __restrict__ stats,
                                     float* __restrict__ d_out, int N) {
  if (threadIdx.x == 0) {
    d_out[3 * (long)N] = stats[36] / (3.0f * (float)N);
    const float r = (sqrtf(stats[37] / (float)N) + sqrtf(stats[38] / (float)N) +
                     sqrtf(stats[39] / (float)N)) / 3.f;
    d_out[3 * (long)N + 1] = r;
  }
}

// ---------------------------------------------------------------------------
// Host-side orchestration
// ---------------------------------------------------------------------------
static inline int capblocks(long total, long cap) {
  long b = (total + 255) / 256;
  if (b > cap) b = cap;
  if (b < 1) b = 1;
  return (int)b;
}

static inline int gemmgrid(int M) {
  return (M + 16 * TILES - 1) / (16 * TILES);
}

static void launch_gemm1(const float* X, const float* W, const float* b,
                         const float* R, float* Y, int M, bool relu,
                         hipStream_t s) {
  dim3 g(gemmgrid(M)), t(256);
  if (R)
    gemm128_wmma<1, false, true><<<g, t, 0, s>>>(
        X, nullptr, nullptr, nullptr, nullptr, nullptr, W, b, R, Y, M);
  else if (relu)
    gemm128_wmma<1, true, false><<<g, t, 0, s>>>(
        X, nullptr, nullptr, nullptr, nullptr, nullptr, W, b, nullptr, Y, M);
  else
    gemm128_wmma<1, false, false><<<g, t, 0, s>>>(
        X, nullptr, nullptr, nullptr, nullptr, nullptr, W, b, nullptr, Y, M);
}

extern "C" void kernel_launch(void* const* d_in, const int* in_sizes, int n_in,
                              void* d_out, int out_size, void* d_ws,
                              size_t ws_size, hipStream_t stream) {
  const float* z = (const float*)d_in[0];
  const float* zt = (const float*)d_in[1];
  const float* pos = (const float*)d_in[2];
  const float* enc_node_W1 = (const float*)d_in[3];
  const float* enc_node_b1 = (const float*)d_in[4];
  const float* enc_node_W2 = (const float*)d_in[5];
  const float* enc_node_b2 = (const float*)d_in[6];
  const float* enc_edge_W1 = (const float*)d_in[7];
  const float* enc_edge_b1 = (const float*)d_in[8];
  const float* enc_edge_W2 = (const float*)d_in[9];
  const float* enc_edge_b2 = (const float*)d_in[10];
  const float* dec_W1 = (const float*)d_in[11];
  const float* dec_b1 = (const float*)d_in[12];
  const float* dec_W2 = (const float*)d_in[13];
  const float* dec_b2 = (const float*)d_in[14];
  const float* beW1 = (const float*)d_in[15];  // [4,384,128]
  const float* beb1 = (const float*)d_in[16];  // [4,128]
  const float* beW2 = (const float*)d_in[17];  // [4,128,128]
  const float* beb2 = (const float*)d_in[18];
  const float* bnW1 = (const float*)d_in[19];  // [4,256,128]
  const float* bnb1 = (const float*)d_in[20];
  const float* bnW2 = (const float*)d_in[21];  // [4,128,128]
  const float* bnb2 = (const float*)d_in[22];
  const int* snd = (const int*)d_in[23];
  const int* rcv = (const int*)d_in[24];

  const int N = in_sizes[1] / 3;
  const int E = in_sizes[23];
  const int T = in_sizes[0] / (N * 3);
  const float* zl = z + (long)(T - 1) * N * 3;  // z[-1]

  // workspace layout (floats)
  float* ws = (float*)d_ws;
  float* stats = ws;                       // 64
  float* ef_raw = stats + 64;              // E*3
  float* ef = ef_raw + (long)E * 3;        // E*128
  float* eh = ef + (long)E * 128;          // E*128
  float* nf = eh + (long)E * 128;          // N*128
  float* nh = nf + (long)N * 128;          // N*128
  float* agg = nh + (long)N * 128;         // N*128
  float* outm = agg + (long)N * 128;       // N*3

  float* zpred = (float*)d_out;

  // --- stats + raw features ---
  fzero_kernel<<<1, 64, 0, stream>>>(stats, 64);
  edge_raw_stats_kernel<<<capblocks(E, 512), 256, 0, stream>>>(pos, snd, rcv,
                                                               ef_raw, stats, E);
  node_stats_kernel<<<capblocks(N, 256), 256, 0, stream>>>(zl, zt, stats, N);
  finalize_stats_kernel<<<1, 16, 0, stream>>>(stats, E, N);

  // --- encoders ---
  enc_l1_kernel<<<capblocks((long)E * 128, 8192), 256, 0, stream>>>(
      ef_raw, stats + 18, stats + 21, enc_edge_W1, enc_edge_b1, eh, E);
  launch_gemm1(eh, enc_edge_W2, enc_edge_b2, nullptr, ef, E, false, stream);
  enc_l1_kernel<<<capblocks((long)N * 128, 8192), 256, 0, stream>>>(
      zl, stats + 24, stats + 27, enc_node_W1, enc_node_b1, nh, N);
  launch_gemm1(nh, enc_node_W2, enc_node_b2, nullptr, nf, N, false, stream);

  // --- message passing blocks ---
  for (int b = 0; b < 4; ++b) {
    const float* eW1 = beW1 + (long)b * 384 * 128;
    const float* eb1 = beb1 + (long)b * 128;
    const float* eW2 = beW2 + (long)b * 128 * 128;
    const float* eb2 = beb2 + (long)b * 128;
    const float* nW1 = bnW1 + (long)b * 256 * 128;
    const float* nb1 = bnb1 + (long)b * 128;
    const float* nW2 = bnW2 + (long)b * 128 * 128;
    const float* nb2 = bnb2 + (long)b * 128;

    // eh = relu([ef | nf[snd] | nf[rcv]] @ eW1 + eb1)
    gemm128_wmma<3, true, false><<<dim3(gemmgrid(E)), 256, 0, stream>>>(
        ef, nullptr, nf, snd, nf, rcv, eW1, eb1, nullptr, eh, E);
    // ef = eh @ eW2 + eb2 + ef   (in-place residual OK: RMW per-thread)
    launch_gemm1(eh, eW2, eb2, ef, ef, E, false, stream);

    // agg = segment_sum(ef, rcv)
    fzero_kernel<<<capblocks((long)N * 128, 8192), 256, 0, stream>>>(
        agg, (long)N * 128);
    scatter_add_kernel<<<capblocks((long)E * 32, 8192), 256, 0, stream>>>(
        ef, rcv, agg, E);

    // nh = relu([nf | agg] @ nW1 + nb1)
    gemm128_wmma<2, true, false><<<dim3(gemmgrid(N)), 256, 0, stream>>>(
        nf, nullptr, agg, nullptr, nullptr, nullptr, nW1, nb1, nullptr, nh, N);
    // nf = nh @ nW2 + nb2 + nf
    launch_gemm1(nh, nW2, nb2, nf, nf, N, false, stream);
  }

  // --- decoder + loss + prediction ---
  launch_gemm1(nf, dec_W1, dec_b1, nullptr, nh, N, true, stream);
  dec_l2_kernel<<<capblocks((long)N * 3, 4096), 256, 0, stream>>>(nh, dec_W2,
                                                                  dec_b2, outm, N);
  loss_pred_kernel<<<capblocks((long)N * 3, 512), 256, 0, stream>>>(
      zl, zt, outm, stats, zpred, stats, N);
  final_scalars_kernel<<<1, 32, 0, stream>>>(stats, zpred, N);
}